// Decoder_88862873354506
// MI455X (gfx1250) — compile-verified
//
#include <hip/hip_runtime.h>
#include <math.h>

#define B_   128
#define P_   196
#define DENC 2048
#define DDEC 512
#define DATT 512
#define DEMB 512
#define V_   10000
#define L_   21
#define T_   20

typedef float v2f __attribute__((ext_vector_type(2)));
typedef float v8f __attribute__((ext_vector_type(8)));

__device__ __forceinline__ float sigmoidf(float x) { return 1.0f / (1.0f + expf(-x)); }

// ---------------------------------------------------------------------------
// Stable descending argsort of caption lengths (B=128, one block).
// ---------------------------------------------------------------------------
__global__ void k_sort(const int* __restrict__ cap_len, const int* __restrict__ caps,
                       int* __restrict__ sind, int* __restrict__ dlens,
                       int* __restrict__ caps_s,
                       float* __restrict__ out_caps, float* __restrict__ out_dlens,
                       float* __restrict__ out_sind)
{
    __shared__ int len_sh[B_];
    __shared__ int sind_sh[B_];
    int i = threadIdx.x;
    len_sh[i] = cap_len[i];
    __syncthreads();
    int li = len_sh[i];
    int rank = 0;
    for (int j = 0; j < B_; ++j) {
        int lj = len_sh[j];
        if (lj > li || (lj == li && j < i)) rank++;
    }
    sind_sh[rank] = i;
    __syncthreads();
    int src = sind_sh[i];
    sind[i] = src;
    out_sind[i] = (float)src;
    int dl = len_sh[src] - 1;
    dlens[i] = dl;
    out_dlens[i] = (float)dl;
    for (int l = 0; l < L_; ++l) {
        int cv = caps[src * L_ + l];
        caps_s[i * L_ + l] = cv;
        out_caps[i * L_ + l] = (float)cv;
    }
}

// ---------------------------------------------------------------------------
// mean_enc[b][d] = mean over p of enc[sort_ind[b]][p][d]
// ---------------------------------------------------------------------------
__global__ void k_mean(const float* __restrict__ enc, const int* __restrict__ sind,
                       float* __restrict__ mean_enc)
{
    int b = blockIdx.x;
    const float* base = enc + (size_t)sind[b] * P_ * DENC;
    for (int d = threadIdx.x; d < DENC; d += blockDim.x) {
        float s = 0.f;
        for (int p = 0; p < P_; ++p) s += base[(size_t)p * DENC + d];
        mean_enc[(size_t)b * DENC + d] = s * (1.0f / (float)P_);
    }
}

// ---------------------------------------------------------------------------
// fp32 WMMA GEMM body: branch-free inner loop. 256 threads = 8 waves; each
// wave computes a full 16x64 tile (4x v_wmma_f32_16x16x4_f32, A frag reused).
// grid = (N/128, M/64).  Optional A row remap, bias, C-accumulate, dlens mask.
// ---------------------------------------------------------------------------
__global__ void k_gemm(const float* __restrict__ A, int lda,
                       const int* __restrict__ sind, int pdim,
                       const float* __restrict__ Bm, int ldb,
                       const float* __restrict__ bias,
                       float* __restrict__ C, long long ldc,
                       int K, int accum,
                       const int* __restrict__ dlens, int t)
{
    const int lane  = threadIdx.x & 31;
    const int wave  = threadIdx.x >> 5;
    const int waveM = wave & 3;         // 4 waves along M  -> 64 rows/block
    const int waveN = wave >> 2;        // 2 waves along N  -> 128 cols/block
    const int m0 = blockIdx.y * 64 + waveM * 16;
    const int n0 = blockIdx.x * 128 + waveN * 64;

    // A fragment: lanes 0-15 hold K=k,k+1 ; lanes 16-31 hold K=k+2,k+3
    const int am   = lane & 15;
    const int koff = (lane >> 4) * 2;
    int arow = m0 + am;
    if (sind) arow = sind[arow / pdim] * pdim + (arow % pdim);
    const float* aptr = A + (size_t)arow * lda + koff;

    const int nn = lane & 15;
    const float* bp0 = Bm + (size_t)koff * ldb + (n0 + nn);
    const float* bp1 = bp0 + 16;
    const float* bp2 = bp0 + 32;
    const float* bp3 = bp0 + 48;

    v8f acc0, acc1, acc2, acc3;
#pragma unroll
    for (int i = 0; i < 8; ++i) { acc0[i] = 0.f; acc1[i] = 0.f; acc2[i] = 0.f; acc3[i] = 0.f; }

#pragma unroll 2
    for (int k = 0; k < K; k += 4) {
        size_t r0 = (size_t)k * ldb;
        size_t r1 = (size_t)(k + 1) * ldb;
        v2f a = *(const v2f*)(aptr + k);
        v2f b0, b1, b2, b3;
        b0.x = bp0[r0]; b0.y = bp0[r1];
        b1.x = bp1[r0]; b1.y = bp1[r1];
        b2.x = bp2[r0]; b2.y = bp2[r1];
        b3.x = bp3[r0]; b3.y = bp3[r1];
        acc0 = __builtin_amdgcn_wmma_f32_16x16x4_f32(false, a, false, b0, (short)0, acc0, false, false);
        acc1 = __builtin_amdgcn_wmma_f32_16x16x4_f32(false, a, false, b1, (short)0, acc1, false, false);
        acc2 = __builtin_amdgcn_wmma_f32_16x16x4_f32(false, a, false, b2, (short)0, acc2, false, false);
        acc3 = __builtin_amdgcn_wmma_f32_16x16x4_f32(false, a, false, b3, (short)0, acc3, false, false);
    }

    // Epilogue: VGPR i -> row m0+i (lanes 0-15) / m0+i+8 (lanes 16-31)
    const int moff = (lane >> 4) * 8;
    v8f* accs[4] = { &acc0, &acc1, &acc2, &acc3 };
#pragma unroll
    for (int j = 0; j < 4; ++j) {
        int col = n0 + 16 * j + nn;
        float bv = bias ? bias[col] : 0.0f;
        v8f av = *accs[j];
#pragma unroll
        for (int i = 0; i < 8; ++i) {
            int row = m0 + moff + i;
            size_t off = (size_t)row * (size_t)ldc + (size_t)col;
            float v = av[i] + bv;
            if (accum) v += C[off];
            if (dlens && t >= dlens[row]) v = 0.0f;
            C[off] = v;
        }
    }
}

// ---------------------------------------------------------------------------
// Edge GEMM: one 16x16 tile per wave (128 threads = 4 waves along M).
// grid = (Nrem/16, M/64); columns start at ncol0.
// ---------------------------------------------------------------------------
__global__ void k_gemm16(const float* __restrict__ A, int lda,
                         const int* __restrict__ sind, int pdim,
                         const float* __restrict__ Bm, int ldb,
                         const float* __restrict__ bias,
                         float* __restrict__ C, long long ldc,
                         int K, int accum,
                         const int* __restrict__ dlens, int t, int ncol0)
{
    const int lane = threadIdx.x & 31;
    const int wave = threadIdx.x >> 5;
    const int m0 = blockIdx.y * 64 + wave * 16;
    const int n0 = ncol0 + blockIdx.x * 16;

    const int am   = lane & 15;
    const int koff = (lane >> 4) * 2;
    int arow = m0 + am;
    if (sind) arow = sind[arow / pdim] * pdim + (arow % pdim);
    const float* aptr = A + (size_t)arow * lda + koff;

    const int nn = lane & 15;
    const float* bp = Bm + (size_t)koff * ldb + (n0 + nn);

    v8f acc;
#pragma unroll
    for (int i = 0; i < 8; ++i) acc[i] = 0.f;

#pragma unroll 4
    for (int k = 0; k < K; k += 4) {
        v2f a = *(const v2f*)(aptr + k);
        v2f b;
        b.x = bp[(size_t)k * ldb];
        b.y = bp[(size_t)(k + 1) * ldb];
        acc = __builtin_amdgcn_wmma_f32_16x16x4_f32(false, a, false, b, (short)0, acc, false, false);
    }

    const int moff = (lane >> 4) * 8;
    int col = n0 + nn;
    float bv = bias ? bias[col] : 0.0f;
#pragma unroll
    for (int i = 0; i < 8; ++i) {
        int row = m0 + moff + i;
        size_t off = (size_t)row * (size_t)ldc + (size_t)col;
        float v = acc[i] + bv;
        if (accum) v += C[off];
        if (dlens && t >= dlens[row]) v = 0.0f;
        C[off] = v;
    }
}

// ---------------------------------------------------------------------------
// e[b,p] = relu(attn1[b,p,:] + attn2[b,:]) . w_f ; alpha = softmax_p(e)
// ---------------------------------------------------------------------------
__global__ void k_attsoftmax(const float* __restrict__ attn1,
                             const float* __restrict__ attn2,
                             const float* __restrict__ wf,
                             const int* __restrict__ dlens, int t,
                             float* __restrict__ alpha,
                             float* __restrict__ alphas_out)
{
    int b = blockIdx.x;
    int lane = threadIdx.x & 31;
    int wave = threadIdx.x >> 5;     // 8 waves
    __shared__ float e_sh[P_];
    __shared__ float mx_sh, sum_sh;

    const float* a2 = attn2 + (size_t)b * DATT;
    for (int p = wave; p < P_; p += 8) {
        const float* a1 = attn1 + ((size_t)b * P_ + p) * DATT;
        float s = 0.f;
        for (int d = lane; d < DATT; d += 32) {
            float v = a1[d] + a2[d];
            v = v > 0.f ? v : 0.f;
            s += v * wf[d];
        }
        for (int off = 16; off; off >>= 1) s += __shfl_xor(s, off, 32);
        if (lane == 0) e_sh[p] = s;
    }
    __syncthreads();

    if (wave == 0) {
        float m = -1e30f;
        for (int p = lane; p < P_; p += 32) m = fmaxf(m, e_sh[p]);
        for (int off = 16; off; off >>= 1) m = fmaxf(m, __shfl_xor(m, off, 32));
        float s = 0.f;
        for (int p = lane; p < P_; p += 32) s += expf(e_sh[p] - m);
        for (int off = 16; off; off >>= 1) s += __shfl_xor(s, off, 32);
        if (lane == 0) { mx_sh = m; sum_sh = s; }
    }
    __syncthreads();

    float m = mx_sh;
    float inv = 1.0f / sum_sh;
    int active = t < dlens[b];
    for (int p = threadIdx.x; p < P_; p += blockDim.x) {
        float av = expf(e_sh[p] - m) * inv;
        alpha[(size_t)b * P_ + p] = av;
        alphas_out[((size_t)b * T_ + t) * P_ + p] = active ? av : 0.0f;
    }
}

// ---------------------------------------------------------------------------
// ctx[b,d] = sum_p enc[sort_ind[b],p,d] * alpha[b,p]
// ---------------------------------------------------------------------------
__global__ void k_ctx(const float* __restrict__ enc, const int* __restrict__ sind,
                      const float* __restrict__ alpha, float* __restrict__ ctx)
{
    int b = blockIdx.x;
    __shared__ float a_sh[P_];
    for (int p = threadIdx.x; p < P_; p += blockDim.x) a_sh[p] = alpha[(size_t)b * P_ + p];
    __syncthreads();
    const float* base = enc + (size_t)sind[b] * P_ * DENC;
    for (int d = threadIdx.x; d < DENC; d += blockDim.x) {
        float s = 0.f;
        for (int p = 0; p < P_; ++p) s += base[(size_t)p * DENC + d] * a_sh[p];
        ctx[(size_t)b * DENC + d] = s;
    }
}

// ---------------------------------------------------------------------------
// x = concat(emb_table[caps_s[:,t]], sigmoid(gatep) * ctx)
// ---------------------------------------------------------------------------
__global__ void k_buildx(const float* __restrict__ emb, const int* __restrict__ caps_s, int t,
                         const float* __restrict__ gatep, const float* __restrict__ ctx,
                         float* __restrict__ x)
{
    int idx = blockIdx.x * blockDim.x + threadIdx.x;
    if (idx >= B_ * (DEMB + DENC)) return;
    int b = idx / (DEMB + DENC);
    int j = idx % (DEMB + DENC);
    float v;
    if (j < DEMB) {
        int tok = caps_s[b * L_ + t];
        v = emb[(size_t)tok * DEMB + j];
    } else {
        int d = j - DEMB;
        v = sigmoidf(gatep[(size_t)b * DENC + d]) * ctx[(size_t)b * DENC + d];
    }
    x[idx] = v;
}

// ---------------------------------------------------------------------------
// LSTM pointwise: gate order i,f,g,o ; masked state update.
// ---------------------------------------------------------------------------
__global__ void k_lstm(const float* __restrict__ g, const int* __restrict__ dlens, int t,
                       float* __restrict__ h, float* __restrict__ c)
{
    int idx = blockIdx.x * blockDim.x + threadIdx.x;
    if (idx >= B_ * DDEC) return;
    int b = idx / DDEC;
    int j = idx % DDEC;
    const float* gb = g + (size_t)b * 4 * DDEC;
    float ig = gb[j];
    float fg = gb[DDEC + j];
    float gg = gb[2 * DDEC + j];
    float og = gb[3 * DDEC + j];
    float cn = sigmoidf(fg) * c[idx] + sigmoidf(ig) * tanhf(gg);
    float hn = sigmoidf(og) * tanhf(cn);
    if (t < dlens[b]) { h[idx] = hn; c[idx] = cn; }
}

// ---------------------------------------------------------------------------
static inline void gemm(const float* A, int lda, const int* sind, int pdim,
                        const float* Bm, int ldb, const float* bias,
                        float* C, long long ldc, int M, int N, int K, int accum,
                        const int* dlens, int t, hipStream_t s)
{
    int nmain = N & ~127;                    // body: multiples of 128 columns
    if (nmain > 0) {
        dim3 grid(nmain / 128, M / 64, 1);
        k_gemm<<<grid, 256, 0, s>>>(A, lda, sind, pdim, Bm, ldb, bias, C, ldc, K, accum, dlens, t);
    }
    int nrem = N - nmain;                    // remainder: 16-col tiles
    if (nrem > 0) {
        dim3 grid(nrem / 16, M / 64, 1);
        k_gemm16<<<grid, 128, 0, s>>>(A, lda, sind, pdim, Bm, ldb, bias, C, ldc, K, accum, dlens, t, nmain);
    }
}

extern "C" void kernel_launch(void* const* d_in, const int* in_sizes, int n_in,
                              void* d_out, int out_size, void* d_ws, size_t ws_size,
                              hipStream_t stream)
{
    const float* enc     = (const float*)d_in[0];
    const int*   caps    = (const int*)d_in[1];
    const int*   cap_len = (const int*)d_in[2];
    const float* W_ea    = (const float*)d_in[3];
    const float* b_ea    = (const float*)d_in[4];
    const float* W_da    = (const float*)d_in[5];
    const float* b_da    = (const float*)d_in[6];
    const float* w_f     = (const float*)d_in[7];
    // d_in[8] = b_f : softmax is shift-invariant, unused
    const float* emb     = (const float*)d_in[9];
    const float* W_ih    = (const float*)d_in[10];
    const float* b_ih    = (const float*)d_in[11];
    const float* W_hh    = (const float*)d_in[12];
    const float* b_hh    = (const float*)d_in[13];
    const float* W_hd    = (const float*)d_in[14];
    const float* b_hd    = (const float*)d_in[15];
    const float* W_cd    = (const float*)d_in[16];
    const float* b_cd    = (const float*)d_in[17];
    const float* W_beta  = (const float*)d_in[18];
    const float* b_beta  = (const float*)d_in[19];
    const float* W_fc    = (const float*)d_in[20];
    const float* b_fc    = (const float*)d_in[21];

    float* out        = (float*)d_out;
    float* out_preds  = out;                                   // B*T*V
    float* out_alphas = out + (size_t)B_ * T_ * V_;            // B*T*P
    float* out_caps   = out_alphas + (size_t)B_ * T_ * P_;     // B*L
    float* out_dlens  = out_caps + (size_t)B_ * L_;            // B
    float* out_sind   = out_dlens + B_;                        // B

    // workspace carve-up (256B aligned)
    char* w = (char*)d_ws;
    size_t cur = 0;
    auto carve = [&](size_t bytes) -> void* {
        void* p = w + cur;
        cur = (cur + bytes + 255) & ~(size_t)255;
        return p;
    };
    int*   sind    = (int*)carve(B_ * sizeof(int));
    int*   dlens   = (int*)carve(B_ * sizeof(int));
    int*   caps_s  = (int*)carve(B_ * L_ * sizeof(int));
    float* mean_e  = (float*)carve((size_t)B_ * DENC * sizeof(float));
    float* h       = (float*)carve((size_t)B_ * DDEC * sizeof(float));
    float* c       = (float*)carve((size_t)B_ * DDEC * sizeof(float));
    float* attn2   = (float*)carve((size_t)B_ * DATT * sizeof(float));
    float* alpha   = (float*)carve((size_t)B_ * P_ * sizeof(float));
    float* ctxb    = (float*)carve((size_t)B_ * DENC * sizeof(float));
    float* gatep   = (float*)carve((size_t)B_ * DENC * sizeof(float));
    float* xbuf    = (float*)carve((size_t)B_ * (DEMB + DENC) * sizeof(float));
    float* g       = (float*)carve((size_t)B_ * 4 * DDEC * sizeof(float));
    float* attn1   = (float*)carve((size_t)B_ * P_ * DATT * sizeof(float)); // 51.4 MB
    (void)ws_size; (void)in_sizes; (void)n_in; (void)out_size;

    // --- preamble ---
    k_sort<<<1, B_, 0, stream>>>(cap_len, caps, sind, dlens, caps_s,
                                 out_caps, out_dlens, out_sind);
    k_mean<<<B_, 256, 0, stream>>>(enc, sind, mean_e);
    // h0 / c0
    gemm(mean_e, DENC, nullptr, 0, W_hd, DDEC, b_hd, h, DDEC, B_, DDEC, DENC, 0, nullptr, 0, stream);
    gemm(mean_e, DENC, nullptr, 0, W_cd, DDEC, b_cd, c, DDEC, B_, DDEC, DENC, 0, nullptr, 0, stream);
    // attn1 = enc_sorted @ W_ea + b_ea   (A gathered via sort_ind row remap)
    gemm(enc, DENC, sind, P_, W_ea, DATT, b_ea, attn1, DATT, B_ * P_, DATT, DENC, 0, nullptr, 0, stream);

    // --- time loop ---
    for (int t = 0; t < T_; ++t) {
        // attn2 = h @ W_da + b_da
        gemm(h, DDEC, nullptr, 0, W_da, DATT, b_da, attn2, DATT, B_, DATT, DDEC, 0, nullptr, 0, stream);
        // e -> softmax -> alpha (+ masked alphas output)
        k_attsoftmax<<<B_, 256, 0, stream>>>(attn1, attn2, w_f, dlens, t, alpha, out_alphas);
        // ctx = alpha^T . enc_sorted
        k_ctx<<<B_, 256, 0, stream>>>(enc, sind, alpha, ctxb);
        // gate pre-activation = h @ W_beta + b_beta
        gemm(h, DDEC, nullptr, 0, W_beta, DENC, b_beta, gatep, DENC, B_, DENC, DDEC, 0, nullptr, 0, stream);
        // x = [emb_t, sigmoid(gatep)*ctx]
        k_buildx<<<(B_ * (DEMB + DENC) + 255) / 256, 256, 0, stream>>>(emb, caps_s, t, gatep, ctxb, xbuf);
        // g = x @ W_ih + b_ih  ;  g += h @ W_hh + b_hh
        gemm(xbuf, DEMB + DENC, nullptr, 0, W_ih, 4 * DDEC, b_ih, g, 4 * DDEC, B_, 4 * DDEC, DEMB + DENC, 0, nullptr, 0, stream);
        gemm(h, DDEC, nullptr, 0, W_hh, 4 * DDEC, b_hh, g, 4 * DDEC, B_, 4 * DDEC, DDEC, 1, nullptr, 0, stream);
        // LSTM state update (masked)
        k_lstm<<<(B_ * DDEC + 255) / 256, 256, 0, stream>>>(g, dlens, t, h, c);
        // preds = h_new @ W_fc + b_fc, masked, written straight into d_out[:, t, :]
        gemm(h, DDEC, nullptr, 0, W_fc, V_, b_fc, out_preds + (size_t)t * V_,
             (long long)T_ * V_, B_, V_, DDEC, 0, dlens, t, stream);
    }
}